// Conv_27693949125154
// MI455X (gfx1250) — compile-verified
//
#include <hip/hip_runtime.h>

// CDNA5 (gfx1250) wave32 WMMA implementation of the fixed-kernel 3x3 conv.
// out[s, 0:84] = relu( x[s, 0:128] . W )  with W a 0/1 matrix (5 ones/col),
// computed exactly in fp32 via V_WMMA_F32_16X16X4_F32.

typedef __attribute__((ext_vector_type(2))) float v2f;
typedef __attribute__((ext_vector_type(8))) float v8f;

#define LDS_STRIDE 132  // 128 K-floats + 4 pad -> conflict-free b64 B loads

__global__ __launch_bounds__(256) void conv_wmma_kernel(
    const float* __restrict__ x, float* __restrict__ out)
{
    // Transposed weight table Bt[n][k] = W[k][n], n in [0,96) (84 real + pad).
    __shared__ float Bt[96 * LDS_STRIDE];

    // Build W once per workgroup: W[k][o]=1 iff dr,dc in [0,2] and (dr+dc) even.
    // 96*128/256 = 48 iterations per thread, uniform (no divergence).
    for (int idx = threadIdx.x; idx < 96 * 128; idx += 256) {
        int n = idx >> 7;       // output column (0..95)
        int k = idx & 127;      // input index   (0..127)
        float v = 0.0f;
        if (n < 84) {
            int r  = n / 6;
            int c  = n - 6 * r;
            int dr = (k >> 3) - r;
            int dc = (k & 7)  - c;
            if (dr >= 0 && dr <= 2 && dc >= 0 && dc <= 2 &&
                (((dr + dc) & 1) == 0))
                v = 1.0f;
        }
        Bt[n * LDS_STRIDE + k] = v;
    }
    __syncthreads();

    const int lane = threadIdx.x & 31;
    const int half = lane >> 4;   // 0: lanes 0-15 (K0/K1), 1: lanes 16-31 (K2/K3)
    const int m    = lane & 15;
    const int gw   = blockIdx.x * 8 + (threadIdx.x >> 5);  // global wave id
    const int s0   = gw * 16;                              // first sample of tile

    // A-fragment base for this lane: row (s0+m), starting at K sub-offset 2*half.
    const float* arow = x + (size_t)(s0 + m) * 128 + 2 * half;

    // Active K-chunk (kc = K/4) masks per N-tile: tile t touches image rows
    // r..r+2 only, so most of the 128-deep K dimension is exactly zero.
    constexpr unsigned ACT[6] = {0x000003FFu, 0x0000FFF0u, 0x000FFC00u,
                                 0x03FF0000u, 0xFFF00000u, 0xFC000000u};

    v8f acc[6];
    #pragma unroll
    for (int t = 0; t < 6; ++t) { v8f z = {}; acc[t] = z; }

    #pragma unroll
    for (int kc = 0; kc < 32; ++kc) {
        // A 16x4 f32 fragment: VGPR0 = K0|K2, VGPR1 = K1|K3 -> one b64/lane.
        v2f a = *(const v2f*)(arow + 4 * kc);
        #pragma unroll
        for (int t = 0; t < 6; ++t) {
            if ((ACT[t] >> kc) & 1u) {   // compile-time after unroll: 60 WMMAs
                const float* bp =
                    &Bt[(16 * t + m) * LDS_STRIDE + 4 * kc + 2 * half];
                v2f b = *(const v2f*)bp;  // ds_load_b64, bank-conflict free
                acc[t] = __builtin_amdgcn_wmma_f32_16x16x4_f32(
                    false, a, false, b, (short)0, acc[t], false, false);
            }
        }
    }

    // D layout: VGPR v -> M = v + 8*half, N = m. Apply ReLU and store.
    #pragma unroll
    for (int t = 0; t < 6; ++t) {
        int o = 16 * t + m;
        if (o < 84) {                       // only t=5 is lane-divergent
            #pragma unroll
            for (int v = 0; v < 8; ++v) {
                int row = s0 + v + 8 * half;
                out[(size_t)row * 84 + o] = fmaxf(acc[t][v], 0.0f);
            }
        }
    }
}

extern "C" void kernel_launch(void* const* d_in, const int* in_sizes, int n_in,
                              void* d_out, int out_size, void* d_ws, size_t ws_size,
                              hipStream_t stream)
{
    const float* x  = (const float*)d_in[0];   // (16384, 14, 128) fp32
    float*      out = (float*)d_out;           // (16384, 14, 84)  fp32

    // 16384*14 = 229376 samples; 16 per wave, 8 waves (256 thr) per block.
    // 229376 / 128 = 1792 blocks, exact -> no tail, EXEC all-1s for WMMA.
    conv_wmma_kernel<<<dim3(1792), dim3(256), 0, stream>>>(x, out);
}